// FuzzyPushdownTransducer_13022340842122
// MI455X (gfx1250) — compile-verified
//
#include <hip/hip_runtime.h>
#include <hip/hip_bf16.h>

typedef __attribute__((ext_vector_type(2))) float v2f;
typedef __attribute__((ext_vector_type(8))) float v8f;

#define NS     64    // states S
#define NI     128   // input symbols I
#define NOUT   64    // output symbols NO
#define KDEP   64    // stack depth K
#define DD     64    // stack value dim D
#define NB     64    // batch B
#define TLEN   256
#define PSTR   0.01f

__device__ __forceinline__ float sigmoidf(float x) { return 1.0f / (1.0f + expf(-x)); }

__global__ __launch_bounds__(1024, 1)
void fpt_kernel(const int* __restrict__ seq,
                const float* __restrict__ Tm,          // (S, I, D, S)
                const float* __restrict__ Om,          // (S, I, D, NO)
                const float* __restrict__ push_gate,   // (S, I, 1)
                const float* __restrict__ pop_gate,    // (S, I, 1)
                const float* __restrict__ push_value,  // (S, I, D)
                const float* __restrict__ init_logits, // (S,)
                float* __restrict__ out)               // (B, T_LEN, NO)
{
    __shared__ float content[KDEP * DD];       // 16 KB stack content
    __shared__ float logits[2][NS][64];        // 32 KB: [0]=out_logits(O), [1]=trans_logits(T)
    __shared__ float state_sh[NS];
    __shared__ float vtop[DD];                 // stack_top
    __shared__ float ptr_sh[KDEP];
    __shared__ float ptr1_sh[KDEP];
    __shared__ float w_sh[KDEP];
    __shared__ float pushv_sh[DD];
    __shared__ float accO[NOUT];
    __shared__ float accN[NS];
    __shared__ float scal[4];                  // [0]=push dot, [1]=pop dot, [2]/[3] init scratch

    const int tid  = threadIdx.x;
    const int lane = tid & 31;
    const int wv   = tid >> 5;                 // 0..31 (wave32)
    const int h    = lane >> 4;                // lane half
    const int n    = lane & 15;
    const int b    = blockIdx.x;

    // ---- init carry: state0 = softmax(init_logits), content0 = 0, pointer0 = e0
    #pragma unroll
    for (int r = 0; r < 4; ++r) content[tid + (r << 10)] = 0.0f;
    if (tid < KDEP) ptr_sh[tid] = (tid == 0) ? 1.0f : 0.0f;
    if (tid == 0) {
        float m = -3.4e38f;
        for (int s = 0; s < NS; ++s) m = fmaxf(m, init_logits[s]);
        float sum = 0.0f;
        for (int s = 0; s < NS; ++s) sum += expf(init_logits[s] - m);
        scal[2] = m; scal[3] = sum;
    }
    __syncthreads();
    if (tid < NS) state_sh[tid] = expf(init_logits[tid] - scal[2]) / scal[3];
    __syncthreads();

    for (int t = 0; t < TLEN; ++t) {
        const int inp = seq[(size_t)b * TLEN + t];

        // ---- P0: zero accumulators   P1: stack_top = pointer @ content
        if (tid < NOUT) { accO[tid] = 0.0f; accN[tid] = 0.0f; }
        if (tid == 0)   { scal[0] = 0.0f; scal[1] = 0.0f; }
        if (tid < DD) {
            float v = 0.0f;
            #pragma unroll 8
            for (int k = 0; k < KDEP; ++k) v += ptr_sh[k] * content[k * DD + tid];
            vtop[tid] = v;
        }
        __syncthreads();

        // ---- P2: big GEMVs via V_WMMA_F32_16X16X4_F32
        // logits[mat][s][j] = sum_d vtop[d] * W[s, d, j],  W = O[:,inp] (mat0) / T[:,inp] (mat1)
        #pragma unroll
        for (int p = 0; p < 4; ++p) {
            const int idx = wv * 4 + p;            // 0..127
            const int mat = idx >> 6;
            const int s   = idx & 63;
            const float* __restrict__ W =
                (mat == 0 ? Om : Tm) + (size_t)(s * NI + inp) * (DD * 64);
            for (int jt = 0; jt < 4; ++jt) {
                v8f c = {0.f, 0.f, 0.f, 0.f, 0.f, 0.f, 0.f, 0.f};
                #pragma unroll 4
                for (int kt = 0; kt < 16; ++kt) {
                    const int k0 = 4 * kt + 2 * h;         // K = r + 2h layout
                    v2f a; a.x = vtop[k0]; a.y = vtop[k0 + 1];
                    v2f bb;
                    bb.x = W[(size_t)k0 * 64 + jt * 16 + n];
                    bb.y = W[(size_t)(k0 + 1) * 64 + jt * 16 + n];
                    c = __builtin_amdgcn_wmma_f32_16x16x4_f32(
                            false, a, false, bb, (short)0, c, false, false);
                }
                if (lane < 16) logits[mat][s][jt * 16 + n] = c[0]; // rows replicated
            }
        }

        // ---- prefetch next symbol's T/O slices into L2 (global_prefetch_b8)
        if (t + 1 < TLEN) {
            const int inpN = seq[(size_t)b * TLEN + t + 1];
            #pragma unroll
            for (int i = 0; i < 8; ++i) {
                const int l    = tid + (i << 10);      // 0..8191 cache lines per matrix
                const int ss   = l >> 7;
                const int line = l & 127;
                const size_t eo = (size_t)(ss * NI + inpN) * 4096 + (size_t)line * 32;
                __builtin_prefetch(Om + eo, 0, 1);
                __builtin_prefetch(Tm + eo, 0, 1);
            }
        }

        // ---- P3: gate dots + push_v (uses OLD state)
        if (tid < NS) {
            const float st = state_sh[tid];
            atomicAdd(&scal[0], st * push_gate[tid * NI + inp]);
            atomicAdd(&scal[1], st * pop_gate[tid * NI + inp]);
        }
        if (tid < DD) {
            float a = 0.0f;
            #pragma unroll 8
            for (int s = 0; s < NS; ++s)
                a += state_sh[s] * push_value[(size_t)(s * NI + inp) * DD + tid];
            pushv_sh[tid] = tanhf(a);
        }
        __syncthreads();

        // ---- P4: row softmax + state-weighted reduction (wave32 shuffles)
        #pragma unroll
        for (int p = 0; p < 4; ++p) {
            const int idx = wv * 4 + p;
            const int mat = idx >> 6;
            const int s   = idx & 63;
            float x0 = logits[mat][s][lane];
            float x1 = logits[mat][s][lane + 32];
            float m = fmaxf(x0, x1);
            for (int off = 16; off > 0; off >>= 1) m = fmaxf(m, __shfl_xor(m, off, 32));
            float e0 = expf(x0 - m), e1 = expf(x1 - m);
            float sum = e0 + e1;
            for (int off = 16; off > 0; off >>= 1) sum += __shfl_xor(sum, off, 32);
            const float w = state_sh[s] / sum;
            float* acc = (mat == 0) ? accO : accN;
            atomicAdd(&acc[lane],      w * e0);
            atomicAdd(&acc[lane + 32], w * e1);
        }
        __syncthreads();

        // ---- P5a: emit output, new state, pop phase of pointer
        const float push_s = sigmoidf(scal[0]);
        const float pop_s  = sigmoidf(scal[1]);
        if (tid < NOUT) out[((size_t)b * TLEN + t) * NOUT + tid] = accO[tid];
        if (tid < NS)   state_sh[tid] = accN[tid];
        if (tid < KDEP) {
            const float pc = ptr_sh[tid];
            const float pn = (tid < KDEP - 1) ? ptr_sh[tid + 1] : 0.0f;
            ptr1_sh[tid] = (pop_s > PSTR) ? (pop_s * pn + (1.0f - pop_s) * pc) : pc;
        }
        __syncthreads();

        // ---- P5b: push phase of pointer
        if (tid < KDEP) {
            const float p1 = ptr1_sh[tid];
            const float pp = (tid > 0) ? ptr1_sh[tid - 1] : 0.0f;
            const float pushed = push_s * pp + (1.0f - push_s) * p1;
            w_sh[tid]   = pushed;
            ptr_sh[tid] = (push_s > PSTR) ? pushed : p1;
        }
        __syncthreads();

        // ---- P5c: content blend on push
        if (push_s > PSTR) {
            #pragma unroll
            for (int r = 0; r < 4; ++r) {
                const int e = tid + (r << 10);
                const int k = e >> 6, d = e & 63;
                const float w = w_sh[k];
                content[e] = content[e] * (1.0f - w) + pushv_sh[d] * w;
            }
        }
        __syncthreads();
    }
}

extern "C" void kernel_launch(void* const* d_in, const int* in_sizes, int n_in,
                              void* d_out, int out_size, void* d_ws, size_t ws_size,
                              hipStream_t stream) {
    (void)in_sizes; (void)n_in; (void)out_size; (void)d_ws; (void)ws_size;
    const int*   seq = (const int*)  d_in[0];
    const float* Tm  = (const float*)d_in[1];
    const float* Om  = (const float*)d_in[2];
    const float* pg  = (const float*)d_in[3];
    const float* qg  = (const float*)d_in[4];
    const float* pv  = (const float*)d_in[5];
    const float* il  = (const float*)d_in[6];
    float* o = (float*)d_out;
    fpt_kernel<<<dim3(NB), dim3(1024), 0, stream>>>(seq, Tm, Om, pg, qg, pv, il, o);
}